// InputsToEmbedding_44676249813596
// MI455X (gfx1250) — compile-verified
//
#include <hip/hip_runtime.h>

// out[b,t,f,e] = x[b,t,f] * W[f,e]   (B*T tokens, F=32, E=16, all f32)
// Memory-bound: 16MB read + 256MB NT-store stream. WMMA f32 16x16x4 with
// zero-padded K computes one 16x16 (tokens x E) tile per feature.

typedef __attribute__((ext_vector_type(2))) float v2f;
typedef __attribute__((ext_vector_type(8))) float v8f;

#define FDIM 32
#define EDIM 16

__global__ __launch_bounds__(256) void bcast_outer_wmma_kernel(
    const float* __restrict__ x,   // [numTokens, FDIM]
    const float* __restrict__ W,   // [FDIM, EDIM]
    float* __restrict__ out,       // [numTokens, FDIM, EDIM]
    int numTokens)
{
    const int lane          = threadIdx.x & 31;
    const int wavesPerBlock = blockDim.x >> 5;
    const int waveId        = blockIdx.x * wavesPerBlock + (threadIdx.x >> 5);
    const int numWaves      = gridDim.x * wavesPerBlock;
    const int tileRows      = numTokens >> 4;   // token groups of 16

    const int  n  = lane & 15;          // E index (and token-in-group for A load)
    const bool lo = lane < 16;

    // Grid-stride over 16-token groups; wave-uniform loop bounds keep EXEC
    // all-1s at every WMMA (ISA requirement).
    for (int tg = waveId; tg < tileRows; tg += numWaves) {
        const int    tok0    = tg << 4;
        const float* xrow    = x   + (size_t)tok0 * FDIM;
        float*       outTile = out + (size_t)tok0 * (FDIM * EDIM);

        for (int f = 0; f < FDIM; ++f) {
            // A[m, K=0] = x[tok0+m, f]; lanes 16-31 (K=2 slot) duplicate lane
            // n's value -> harmless: B rows 1..3 are zero. VGPR1 (K=1/3) = 0.
            float a0 = xrow[(size_t)n * FDIM + f];
            // B[K=0, n] = W[f, n] in lanes 0-15; zero elsewhere (v_cndmask,
            // no EXEC manipulation).
            float bw = W[f * EDIM + n];
            float b0 = lo ? bw : 0.0f;

            v2f A = {a0, 0.0f};
            v2f B = {b0, 0.0f};
            v8f C = {};
            // D[m,n] = sum_k A[m,k]*B[k,n] = x[tok0+m,f] * W[f,n]
            v8f D = __builtin_amdgcn_wmma_f32_16x16x4_f32(
                /*neg_a=*/false, A, /*neg_b=*/false, B,
                /*c_mod=*/(short)0, C, /*reuse_a=*/false, /*reuse_b=*/false);

            // D VGPR r: lanes 0-15 -> row m=r, lanes 16-31 -> row m=8+r.
            const int mBase = (lane >> 4) << 3;     // 0 or 8
            float* o = outTile + (size_t)f * EDIM + n;
            #pragma unroll
            for (int r = 0; r < 8; ++r) {
                // write-once 256MB stream: non-temporal store (TH=NT)
                __builtin_nontemporal_store(D[r],
                    o + (size_t)(mBase + r) * (FDIM * EDIM));
            }
        }
    }
}

extern "C" void kernel_launch(void* const* d_in, const int* in_sizes, int n_in,
                              void* d_out, int out_size, void* d_ws, size_t ws_size,
                              hipStream_t stream) {
    const float* x = (const float*)d_in[0];   // [B*T, F] f32
    const float* W = (const float*)d_in[1];   // [F, E]   f32
    float* out = (float*)d_out;               // [B*T, F, E] f32

    const int numTokens = in_sizes[0] / FDIM; // B*T = 131072
    const int tileRows  = numTokens >> 4;     // 8192 groups of 16 tokens

    const int threads       = 256;            // 8 wave32 per block
    const int wavesPerBlock = threads / 32;
    int blocks = (tileRows + wavesPerBlock - 1) / wavesPerBlock;  // 1024
    if (blocks > 4096) blocks = 4096;
    if (blocks < 1)    blocks = 1;

    bcast_outer_wmma_kernel<<<blocks, threads, 0, stream>>>(x, W, out, numTokens);
}